// StackedAttention_16836271800801
// MI455X (gfx1250) — compile-verified
//
#include <hip/hip_runtime.h>
#include <hip/hip_bf16.h>
#include <math.h>

// Stacked cross attention (i2t), f32. Strategy:
//   S[w,r] = cap[c,w].img[i,r]  via V_WMMA_F32_16X16X4_F32 (exact f32 path)
//   w12[r] = sum_w a2[r,w]*S[w,r]            (reuse scores, no 2nd big GEMM)
//   |wctx_r|^2 = a2_r^T G_c a2_r             (precomputed 50x50 Gram per caption)

typedef __attribute__((ext_vector_type(2))) float v2f;
typedef __attribute__((ext_vector_type(8))) float v8f;

#define N_IMG   64
#define N_REG   36
#define N_CAP   64
#define N_WRD   50
#define DIM     1024
#define KC      32      // K chunk staged in LDS
#define CAP_LD  36      // LDS row stride (floats): 144B, 16B aligned, conflict-free
#define S_LD    49      // score-tile row stride (odd -> conflict-free col access)
#define SMOOTH  4.0f
#define EPSV    1e-8f

// ---------------------------------------------------------------- Gram kernel
// G_c[w,wp] = captions[c,w] . captions[c,wp]   (50x50 padded to 64x64 tiles)
__global__ __launch_bounds__(128) void StackedAttn_gram(
    const float* __restrict__ captions, float* __restrict__ gram)
{
    __shared__ float capL[64 * CAP_LD];
    const int c    = blockIdx.x;
    const int tid  = threadIdx.x;
    const int wave = tid >> 5, lane = tid & 31;
    const int half = lane >> 4, m = lane & 15;

    // zero pad rows 50..63 once
    for (int idx = tid; idx < 14 * CAP_LD; idx += 128) capL[50 * CAP_LD + idx] = 0.f;

    v8f acc0 = {}, acc1 = {}, acc2 = {}, acc3 = {};
    const float* capBase = captions + (size_t)c * N_WRD * DIM;

    for (int k0 = 0; k0 < DIM; k0 += KC) {
        __syncthreads();
        {   // 50 rows x 8 float4 segments
            int seg = tid & 7;
            for (int row = tid >> 3; row < N_WRD; row += 16) {
                float4 v = *(const float4*)(capBase + row * DIM + k0 + seg * 4);
                *(float4*)(capL + row * CAP_LD + seg * 4) = v;
            }
        }
        __syncthreads();
        const float* aP = capL + ((wave << 4) + m) * CAP_LD + 2 * half;
        const float* b0 = capL + (m) * CAP_LD + 2 * half;
        const float* b1 = capL + (16 + m) * CAP_LD + 2 * half;
        const float* b2 = capL + (32 + m) * CAP_LD + 2 * half;
        const float* b3 = capL + (48 + m) * CAP_LD + 2 * half;
        #pragma unroll
        for (int kk = 0; kk < KC; kk += 4) {
            v2f a  = *(const v2f*)(aP + kk);
            v2f v0 = *(const v2f*)(b0 + kk);
            v2f v1 = *(const v2f*)(b1 + kk);
            v2f v2 = *(const v2f*)(b2 + kk);
            v2f v3 = *(const v2f*)(b3 + kk);
            acc0 = __builtin_amdgcn_wmma_f32_16x16x4_f32(false, a, false, v0, (short)0, acc0, false, false);
            acc1 = __builtin_amdgcn_wmma_f32_16x16x4_f32(false, a, false, v1, (short)0, acc1, false, false);
            acc2 = __builtin_amdgcn_wmma_f32_16x16x4_f32(false, a, false, v2, (short)0, acc2, false, false);
            acc3 = __builtin_amdgcn_wmma_f32_16x16x4_f32(false, a, false, v3, (short)0, acc3, false, false);
        }
    }
    float* gOut = gram + (size_t)c * N_WRD * N_WRD;
    #pragma unroll
    for (int v = 0; v < 8; ++v) {
        int w = (wave << 4) + v + (half << 3);
        if (w < N_WRD) {
            if (m      < N_WRD) gOut[w * N_WRD + m]      = acc0[v];
            if (16 + m < N_WRD) gOut[w * N_WRD + 16 + m] = acc1[v];
            if (32 + m < N_WRD) gOut[w * N_WRD + 32 + m] = acc2[v];
            if (48 + m < N_WRD) gOut[w * N_WRD + 48 + m] = acc3[v];
        }
    }
}

// ---------------------------------------------------------------- region norms
__global__ __launch_bounds__(64) void StackedAttn_norms(
    const float* __restrict__ images, float* __restrict__ n1)
{
    const int i = blockIdx.x, t = threadIdx.x;
    if (t < N_REG) {
        const float4* p = (const float4*)(images + ((size_t)i * N_REG + t) * DIM);
        float s = 0.f;
        #pragma unroll 4
        for (int d = 0; d < DIM / 4; ++d) {
            float4 v = p[d];
            s += v.x * v.x + v.y * v.y + v.z * v.z + v.w * v.w;
        }
        n1[i * N_REG + t] = sqrtf(s);
    }
}

// ---------------------------------------------------------------- main kernel
__global__ __launch_bounds__(128) void StackedAttn_main(
    const float* __restrict__ images,    // (64,36,1024)
    const float* __restrict__ captions,  // (64,50,1024)
    const int*   __restrict__ cap_lens,  // (64)
    const float* __restrict__ gram,      // (64,50,50)
    const float* __restrict__ n1,        // (64,36)
    float*       __restrict__ out)       // (64,64): out[i*64 + c]
{
    __shared__ float Sbuf[64 * S_LD];                       // scores 64x48 (pad 49)
    __shared__ float A1[64 * S_LD];                         // region-softmax
    __shared__ float chunk[(64 + 48) * CAP_LD];             // cap+img K-chunks; reused as A2
    __shared__ float Gbuf[N_WRD * N_WRD];                   // caption Gram
    __shared__ float red[N_REG];

    const int c = blockIdx.x, i = blockIdx.y;
    const int tid  = threadIdx.x;
    const int wave = tid >> 5, lane = tid & 31;
    const int half = lane >> 4, m = lane & 15;

    float* capL = chunk;
    float* imgL = chunk + 64 * CAP_LD;

    // zero pad rows once (rewritten rows < 50 / < 36 every chunk; pads persist)
    for (int idx = tid; idx < 14 * CAP_LD; idx += 128) capL[50 * CAP_LD + idx] = 0.f;
    for (int idx = tid; idx < 12 * CAP_LD; idx += 128) imgL[36 * CAP_LD + idx] = 0.f;

    v8f acc0 = {}, acc1 = {}, acc2 = {};
    const float* capBase = captions + (size_t)c * N_WRD * DIM;
    const float* imgBase = images   + (size_t)i * N_REG * DIM;

    for (int k0 = 0; k0 < DIM; k0 += KC) {
        __syncthreads();
        {   // stage caption chunk: 50 rows x 8 float4
            int seg = tid & 7;
            for (int row = tid >> 3; row < N_WRD; row += 16) {
                float4 v = *(const float4*)(capBase + row * DIM + k0 + seg * 4);
                *(float4*)(capL + row * CAP_LD + seg * 4) = v;
            }
            // stage image chunk: 36 rows x 8 float4
            for (int row = tid >> 3; row < N_REG; row += 16) {
                float4 v = *(const float4*)(imgBase + row * DIM + k0 + seg * 4);
                *(float4*)(imgL + row * CAP_LD + seg * 4) = v;
            }
        }
        __syncthreads();
        // A frag: lanes 0-15 hold row M, K={k,k+1}; lanes 16-31 K={k+2,k+3}.
        // B frag mirrors with N across lanes (B[k][r] = img[r][k]).
        const float* aP = capL + ((wave << 4) + m) * CAP_LD + 2 * half;
        const float* b0 = imgL + (m) * CAP_LD + 2 * half;
        const float* b1 = imgL + (16 + m) * CAP_LD + 2 * half;
        const float* b2 = imgL + (32 + m) * CAP_LD + 2 * half;
        #pragma unroll
        for (int kk = 0; kk < KC; kk += 4) {
            v2f a  = *(const v2f*)(aP + kk);
            v2f v0 = *(const v2f*)(b0 + kk);
            v2f v1 = *(const v2f*)(b1 + kk);
            v2f v2 = *(const v2f*)(b2 + kk);
            acc0 = __builtin_amdgcn_wmma_f32_16x16x4_f32(false, a, false, v0, (short)0, acc0, false, false);
            acc1 = __builtin_amdgcn_wmma_f32_16x16x4_f32(false, a, false, v1, (short)0, acc1, false, false);
            acc2 = __builtin_amdgcn_wmma_f32_16x16x4_f32(false, a, false, v2, (short)0, acc2, false, false);
        }
    }

    // scatter accumulators to LDS (C layout: VGPR v -> row v / v+8, col = lane%16)
    #pragma unroll
    for (int v = 0; v < 8; ++v) {
        int w = (wave << 4) + v + (half << 3);
        Sbuf[w * S_LD + m]      = acc0[v];
        Sbuf[w * S_LD + 16 + m] = acc1[v];
        Sbuf[w * S_LD + 32 + m] = acc2[v];
    }
    // stage Gram while scores settle
    for (int idx = tid; idx < N_WRD * N_WRD; idx += 128)
        Gbuf[idx] = gram[(size_t)c * N_WRD * N_WRD + idx];
    __syncthreads();

    // softmax over regions r, per word row w
    if (tid < 64) {
        const int w = tid;
        float mx = -3.4e38f;
        for (int r = 0; r < N_REG; ++r) mx = fmaxf(mx, Sbuf[w * S_LD + r]);
        float s = 0.f;
        for (int r = 0; r < N_REG; ++r) {
            float e = __expf(Sbuf[w * S_LD + r] - mx);
            A1[w * S_LD + r] = e; s += e;
        }
        float inv = 1.f / s;
        for (int r = 0; r < N_REG; ++r) A1[w * S_LD + r] *= inv;
    }
    __syncthreads();

    // masked smooth softmax over words, then w12 and Gram-based norm
    float* A2 = chunk;                 // overlay: [36][52]
    const int len = cap_lens[c];
    if (tid < N_REG) {
        const int r = tid;
        float mx = -3.4e38f;
        for (int w = 0; w < len; ++w) mx = fmaxf(mx, SMOOTH * A1[w * S_LD + r]);
        float s = 0.f;
        for (int w = 0; w < len; ++w) {
            float e = __expf(SMOOTH * A1[w * S_LD + r] - mx);
            A2[r * 52 + w] = e; s += e;
        }
        float inv = 1.f / s, w12 = 0.f;
        for (int w = 0; w < len; ++w) {
            float a2v = A2[r * 52 + w] * inv;
            A2[r * 52 + w] = a2v;
            w12 += a2v * Sbuf[w * S_LD + r];
        }
        float n2sq = 0.f;
        for (int w = 0; w < len; ++w) {
            const float* Grow = Gbuf + w * N_WRD;
            float gv = 0.f;
            for (int wp = 0; wp < len; ++wp) gv += Grow[wp] * A2[r * 52 + wp];
            n2sq += A2[r * 52 + w] * gv;
        }
        float n2 = sqrtf(fmaxf(n2sq, 0.f));
        red[r] = w12 / fmaxf(n1[i * N_REG + r] * n2, EPSV);
    }
    __syncthreads();
    if (tid == 0) {
        float s = 0.f;
        for (int r = 0; r < N_REG; ++r) s += red[r];
        out[i * N_CAP + c] = s * (1.0f / N_REG);
    }
}

extern "C" void kernel_launch(void* const* d_in, const int* in_sizes, int n_in,
                              void* d_out, int out_size, void* d_ws, size_t ws_size,
                              hipStream_t stream) {
    const float* images   = (const float*)d_in[0];   // 64*36*1024
    const float* captions = (const float*)d_in[1];   // 64*50*1024
    const int*   cap_lens = (const int*)d_in[2];     // 64
    float* out  = (float*)d_out;                     // 64*64 (i,c)
    float* gram = (float*)d_ws;                      // 64*50*50 floats
    float* n1   = gram + (size_t)N_CAP * N_WRD * N_WRD;  // 64*36 floats

    StackedAttn_gram <<<N_CAP, 128, 0, stream>>>(captions, gram);
    StackedAttn_norms<<<N_IMG, 64, 0, stream>>>(images, n1);
    dim3 grid(N_CAP, N_IMG);
    StackedAttn_main <<<grid, 128, 0, stream>>>(images, captions, cap_lens, gram, n1, out);
}